// InfoNCELoss_50087908606088
// MI455X (gfx1250) — compile-verified
//
#include <hip/hip_runtime.h>
#include <hip/hip_bf16.h>

// ---------------------------------------------------------------------------
// InfoNCE loss over per-image L2 norms.
// Bandwidth-bound: 77 MB streamed once -> ~3.3 us floor at 23.3 TB/s.
// Stage 1 uses V_WMMA_F32_16X16X4_F32 as an exact fp32 32-lane reducer.
// ---------------------------------------------------------------------------

#define IMGS_PER_SIDE 64
#define N_IMGS        128                      // 64 pos + 64 neg
#define PER_IMG       (3 * 224 * 224)          // 150528 floats
#define VEC_PER_IMG   (PER_IMG / 4)            // 37632 float4
#define NB            21                       // blocks per image
#define VEC_PER_BLK   (VEC_PER_IMG / NB)       // 1792 float4
#define THREADS       256
#define VEC_PER_THR   (VEC_PER_BLK / THREADS)  // 7 float4, exact
#define TEMPERATURE   0.7f

typedef __attribute__((ext_vector_type(2))) float v2f;
typedef __attribute__((ext_vector_type(8))) float v8f;

// Exact fp32 reduction of 32 lane values via one WMMA:
//   A (16x4): VGPR0 = acc (one value per lane), VGPR1 = 0
//   B (4x16): all ones (layout-independent since constant)
//   D[m][n] = acc[m] + acc[m+16] for every n.
// Per-lane sum of D's 8 VGPRs covers rows 0-7 (lanes 0-15) / 8-15 (lanes
// 16-31); shfl_xor(16)-add yields the full wave sum in every lane.
__device__ __forceinline__ float wave_reduce_wmma(float acc) {
    v2f a; a[0] = acc;  a[1] = 0.0f;
    v2f b; b[0] = 1.0f; b[1] = 1.0f;
    v8f c = {};
    v8f d = __builtin_amdgcn_wmma_f32_16x16x4_f32(
        /*neg_a=*/false, a, /*neg_b=*/false, b,
        /*c_mod=*/(short)0, c, /*reuse_a=*/false, /*reuse_b=*/false);
    float s = d[0] + d[1] + d[2] + d[3] + d[4] + d[5] + d[6] + d[7];
    s += __shfl_xor(s, 16, 32);
    return s;
}

// Stage 1: one partial sum-of-squares per block. Grid = N_IMGS * NB blocks.
__global__ __launch_bounds__(THREADS)
void infonce_partial_sumsq(const float* __restrict__ pos,
                           const float* __restrict__ neg,
                           float* __restrict__ partials) {
    const int img   = blockIdx.x / NB;   // 0..127
    const int chunk = blockIdx.x % NB;   // 0..20

    const float* src = (img < IMGS_PER_SIDE)
                           ? pos + (size_t)img * PER_IMG
                           : neg + (size_t)(img - IMGS_PER_SIDE) * PER_IMG;
    const float4* p = reinterpret_cast<const float4*>(src)
                      + (size_t)chunk * VEC_PER_BLK + threadIdx.x;

    float acc = 0.0f;
#pragma unroll
    for (int i = 0; i < VEC_PER_THR; ++i) {
        float4 v = p[i * THREADS];       // coalesced 4 KB per iteration
        acc = fmaf(v.x, v.x, acc);
        acc = fmaf(v.y, v.y, acc);
        acc = fmaf(v.z, v.z, acc);
        acc = fmaf(v.w, v.w, acc);
    }

    // Wave reduction via WMMA (EXEC all-1s here: no divergence).
    float wtot = wave_reduce_wmma(acc);

    __shared__ float wsum[THREADS / 32];
    const int wave = threadIdx.x >> 5;
    const int lane = threadIdx.x & 31;
    if (lane == 0) wsum[wave] = wtot;
    __syncthreads();

    if (threadIdx.x == 0) {
        float t = 0.0f;
#pragma unroll
        for (int i = 0; i < THREADS / 32; ++i) t += wsum[i];
        partials[blockIdx.x] = t;        // index == img*NB + chunk
    }
}

// Stage 2: single block of 128 threads. Deterministic serial reductions
// over the 128 norms (trivial work).
__global__ __launch_bounds__(N_IMGS)
void infonce_finalize(const float* __restrict__ partials,
                      float* __restrict__ out) {
    __shared__ float norms[N_IMGS];
    __shared__ float ex[N_IMGS];
    __shared__ float stats[2];           // mu, std*T

    const int t = threadIdx.x;           // 0..127 == image id
    float s = 0.0f;
#pragma unroll
    for (int j = 0; j < NB; ++j) s += partials[t * NB + j];
    const float norm = sqrtf(s);
    norms[t] = norm;
    __syncthreads();

    if (t == 0) {
        float m = 0.0f;
        for (int i = 0; i < N_IMGS; ++i) m += norms[i];
        m *= (1.0f / N_IMGS);
        float v = 0.0f;
        for (int i = 0; i < N_IMGS; ++i) {
            float dd = norms[i] - m;
            v = fmaf(dd, dd, v);
        }
        v *= (1.0f / N_IMGS);            // ddof = 0, matches jnp.std
        stats[0] = m;
        stats[1] = sqrtf(v) * TEMPERATURE;
    }
    __syncthreads();

    ex[t] = expf((norm - stats[0]) / stats[1]);
    __syncthreads();

    if (t == 0) {
        float ps = 0.0f, ns = 0.0f;
        for (int i = 0; i < IMGS_PER_SIDE; ++i)      ps += ex[i];
        for (int i = IMGS_PER_SIDE; i < N_IMGS; ++i) ns += ex[i];
        out[0] = -logf(ps / (ps + ns));
    }
}

extern "C" void kernel_launch(void* const* d_in, const int* in_sizes, int n_in,
                              void* d_out, int out_size, void* d_ws, size_t ws_size,
                              hipStream_t stream) {
    const float* pos = (const float*)d_in[0];   // [64,3,224,224] fp32
    const float* neg = (const float*)d_in[1];   // [64,3,224,224] fp32
    float* partials  = (float*)d_ws;            // N_IMGS*NB floats, fully
                                                // rewritten every call
    float* out = (float*)d_out;

    infonce_partial_sumsq<<<N_IMGS * NB, THREADS, 0, stream>>>(pos, neg, partials);
    infonce_finalize<<<1, N_IMGS, 0, stream>>>(partials, out);
}